// FrequencySemanticCipherOptimized_7370163880081
// MI455X (gfx1250) — compile-verified
//
#include <hip/hip_runtime.h>
#include <stdint.h>

#define EPS 1e-8f

// ---------- helpers ----------------------------------------------------------

// Monotonic float <-> uint mapping so unsigned atomicMin/Max order like floats.
__device__ __forceinline__ unsigned fenc(float f) {
    unsigned u = __float_as_uint(f);
    return u ^ (unsigned)(((int)u >> 31) | 0x80000000u);
}
__device__ __forceinline__ float fdec(unsigned e) {
    unsigned u = (e & 0x80000000u) ? (e ^ 0x80000000u) : ~e;
    return __uint_as_float(u);
}

// CDNA5 async global->LDS copies (tracked by ASYNCcnt). GVS addressing:
// mem = SGPR base + VGPR 32-bit offset ; LDS dest byte address in VGPR.
__device__ __forceinline__ void async_ld_b128(unsigned lds, unsigned voff, const void* base) {
    asm volatile("global_load_async_to_lds_b128 %0, %1, %2"
                 :: "v"(lds), "v"(voff), "s"(base) : "memory");
}
__device__ __forceinline__ void async_ld_b64(unsigned lds, unsigned voff, const void* base) {
    asm volatile("global_load_async_to_lds_b64 %0, %1, %2"
                 :: "v"(lds), "v"(voff), "s"(base) : "memory");
}

// ---------- workspace layout (unsigned[36]) ----------------------------------
// ws[0..3]   : max|noise| bits (plain bits; values are >= 0)
// ws[4..19]  : per-batch min, fenc-encoded
// ws[20..35] : per-batch max, fenc-encoded

__global__ void fsc_init_ws(unsigned* ws) {
    int t = threadIdx.x;
    if (t < 4)        ws[t] = 0u;            // abs-max accumulators
    else if (t < 20)  ws[t] = 0xFFFFFFFFu;   // min accumulators (encoded +max)
    else if (t < 36)  ws[t] = 0u;            // max accumulators (encoded -max)
}

// ---------- pass 1: global max|n| per noise tensor ---------------------------

__global__ __launch_bounds__(256)
void fsc_noise_absmax(const float4* __restrict__ p0, const float4* __restrict__ p1,
                      const float4* __restrict__ p2, const float4* __restrict__ p3,
                      unsigned* __restrict__ ws, int n4) {
    __shared__ float red[256];
    const float4* p = (blockIdx.y == 0) ? p0 : (blockIdx.y == 1) ? p1
                    : (blockIdx.y == 2) ? p2 : p3;
    float m = 0.0f;
    for (size_t i = (size_t)blockIdx.x * 256 + threadIdx.x; i < (size_t)n4;
         i += (size_t)gridDim.x * 256) {
        float4 v = p[i];
        m = fmaxf(m, fmaxf(fmaxf(fabsf(v.x), fabsf(v.y)),
                           fmaxf(fabsf(v.z), fabsf(v.w))));
    }
    red[threadIdx.x] = m;
    __syncthreads();
    for (int s = 128; s > 0; s >>= 1) {
        if (threadIdx.x < s)
            red[threadIdx.x] = fmaxf(red[threadIdx.x], red[threadIdx.x + s]);
        __syncthreads();
    }
    if (threadIdx.x == 0)
        atomicMax(&ws[blockIdx.y], __float_as_uint(red[0]));  // values >= 0
}

// ---------- passes 2 & 3: fused haar-cipher (reduce / write) -----------------
// B=16, C=3, H=W=1024, Hh=Wh=512. blockIdx.y = batch. Each block iterates over
// half-rows (C*Hh = 1536 per batch) with a double-buffered async global->LDS
// pipeline: while tile i is computed from buffer p, tile i+1 streams into
// buffer p^1 (ASYNCcnt <= 6 separates the two in-order groups of 6 ops).
// Thread t owns half-positions 2t and 2t+1 -> b128 LDS reads, float4 stores.

template <bool WRITE>
__global__ __launch_bounds__(256)
void fsc_cipher(const float* __restrict__ img,
                const float* __restrict__ nLL, const float* __restrict__ nLH,
                const float* __restrict__ nHL, const float* __restrict__ nHH,
                float* __restrict__ out, unsigned* __restrict__ ws) {
    __shared__ alignas(16) float sImg[2][2048];   // per buf: rows 2y, 2y+1
    __shared__ alignas(16) float sNs[2][2048];    // per buf: LL|LH|HL|HH x 512
    __shared__ float redMin[256];
    __shared__ float redMax[256];

    const int t = threadIdx.x;
    const int b = blockIdx.y;

    // Noise-normalization coefficients (include the 0.5 from the iDWT and
    // ALPHA_LOW=0.2 / ALPHA_HIGH=1.0).
    const float kLL = 0.1f / (__uint_as_float(ws[0]) + EPS);
    const float kLH = 0.5f / (__uint_as_float(ws[1]) + EPS);
    const float kHL = 0.5f / (__uint_as_float(ws[2]) + EPS);
    const float kHH = 0.5f / (__uint_as_float(ws[3]) + EPS);

    float mn = 0.0f, inv = 1.0f;
    if (WRITE) {
        mn = fdec(ws[4 + b]);
        float mx = fdec(ws[20 + b]);
        inv = 1.0f / fmaxf(mx - mn, EPS);
    }

    const unsigned ldsImg = (unsigned)(uintptr_t)&sImg[0][0];
    const unsigned ldsN   = (unsigned)(uintptr_t)&sNs[0][0];

    auto issue = [&](int rr, int buf) {
        const int c  = rr >> 9;
        const int yh = rr & 511;
        const size_t chan = (size_t)(b * 3 + c);
        const size_t row0 = (chan << 20) + ((size_t)(2 * yh) << 10);  // elems
        const size_t nb   = (chan << 18) + ((size_t)yh << 9);         // elems
        const unsigned ib  = ldsImg + (unsigned)buf * 8192u;
        const unsigned nbs = ldsN   + (unsigned)buf * 8192u;
        async_ld_b128(ib +          16u * t, 16u * t, img + row0);
        async_ld_b128(ib + 4096u +  16u * t, 16u * t, img + row0 + 1024);
        async_ld_b64 (nbs +          8u * t,  8u * t, nLL + nb);
        async_ld_b64 (nbs + 2048u +  8u * t,  8u * t, nLH + nb);
        async_ld_b64 (nbs + 4096u +  8u * t,  8u * t, nHL + nb);
        async_ld_b64 (nbs + 6144u +  8u * t,  8u * t, nHH + nb);
    };

    float locMin = 3.4e38f, locMax = -3.4e38f;

    int pb = 0;
    issue(blockIdx.x, 0);  // prologue: tile 0 -> buffer 0

    for (int r = blockIdx.x; r < 1536; r += gridDim.x, pb ^= 1) {
        const int  rn      = r + (int)gridDim.x;
        const bool hasNext = rn < 1536;
        if (hasNext) {
            issue(rn, pb ^ 1);  // prefetch next tile into other buffer
            // Async loads complete in order: <=6 outstanding => current tile
            // (the older group of 6) has fully landed in LDS.
            asm volatile("s_wait_asynccnt 0x6" ::: "memory");
        } else {
            asm volatile("s_wait_asynccnt 0x0" ::: "memory");
        }
        __syncthreads();  // all waves' loads for this tile have landed

        const float* sI = &sImg[pb][0];
        const float* sN = &sNs[pb][0];
        const float4 ab  = ((const float4*)sI)[t];            // a0 b0 a1 b1
        const float4 cd  = ((const float4*)(sI + 1024))[t];   // c0 d0 c1 d1
        const float2 vll = ((const float2*)sN)[t];
        const float2 vlh = ((const float2*)(sN + 512))[t];
        const float2 vhl = ((const float2*)(sN + 1024))[t];
        const float2 vhh = ((const float2*)(sN + 1536))[t];

        const float s00 = kLL * vll.x, s10 = kLH * vlh.x,
                    s20 = kHL * vhl.x, s30 = kHH * vhh.x;
        const float s01 = kLL * vll.y, s11 = kLH * vlh.y,
                    s21 = kHL * vhl.y, s31 = kHH * vhh.y;

        const float pa0 = ab.x + (s00 + s10 + s20 + s30);
        const float pb0 = ab.y + (s00 + s10 - s20 - s30);
        const float pc0 = cd.x + (s00 - s10 + s20 - s30);
        const float pd0 = cd.y + (s00 - s10 - s20 + s30);
        const float pa1 = ab.z + (s01 + s11 + s21 + s31);
        const float pb1 = ab.w + (s01 + s11 - s21 - s31);
        const float pc1 = cd.z + (s01 - s11 + s21 - s31);
        const float pd1 = cd.w + (s01 - s11 - s21 + s31);

        if (WRITE) {
            const int c  = r >> 9;
            const int yh = r & 511;
            const size_t chan = (size_t)(b * 3 + c);
            const size_t row0 = (chan << 20) + ((size_t)(2 * yh) << 10);
            float4* o0 = (float4*)(out + row0);
            float4* o1 = (float4*)(out + row0 + 1024);
            o0[t] = make_float4((pa0 - mn) * inv, (pb0 - mn) * inv,
                                (pa1 - mn) * inv, (pb1 - mn) * inv);
            o1[t] = make_float4((pc0 - mn) * inv, (pd0 - mn) * inv,
                                (pc1 - mn) * inv, (pd1 - mn) * inv);
        } else {
            locMin = fminf(locMin,
                           fminf(fminf(fminf(pa0, pb0), fminf(pc0, pd0)),
                                 fminf(fminf(pa1, pb1), fminf(pc1, pd1))));
            locMax = fmaxf(locMax,
                           fmaxf(fmaxf(fmaxf(pa0, pb0), fmaxf(pc0, pd0)),
                                 fmaxf(fmaxf(pa1, pb1), fmaxf(pc1, pd1))));
        }
        __syncthreads();  // buffer pb consumed; safe target 2 iterations later
    }

    if (!WRITE) {
        redMin[t] = locMin;
        redMax[t] = locMax;
        __syncthreads();
        for (int s = 128; s > 0; s >>= 1) {
            if (t < s) {
                redMin[t] = fminf(redMin[t], redMin[t + s]);
                redMax[t] = fmaxf(redMax[t], redMax[t + s]);
            }
            __syncthreads();
        }
        if (t == 0) {
            atomicMin(&ws[4 + b],  fenc(redMin[0]));
            atomicMax(&ws[20 + b], fenc(redMax[0]));
        }
    }
}

// ---------- launch -----------------------------------------------------------

extern "C" void kernel_launch(void* const* d_in, const int* in_sizes, int n_in,
                              void* d_out, int out_size, void* d_ws, size_t ws_size,
                              hipStream_t stream) {
    const float* img = (const float*)d_in[0];
    const float* nLL = (const float*)d_in[1];
    const float* nLH = (const float*)d_in[2];
    const float* nHL = (const float*)d_in[3];
    const float* nHH = (const float*)d_in[4];
    float* out = (float*)d_out;
    unsigned* ws = (unsigned*)d_ws;

    const int n4 = in_sizes[1] / 4;  // 12,582,912 / 4 per noise tensor

    fsc_init_ws<<<1, 64, 0, stream>>>(ws);

    fsc_noise_absmax<<<dim3(1024, 4), 256, 0, stream>>>(
        (const float4*)nLL, (const float4*)nLH,
        (const float4*)nHL, (const float4*)nHH, ws, n4);

    // pass 2: per-batch min/max of encrypted image
    fsc_cipher<false><<<dim3(192, 16), 256, 0, stream>>>(
        img, nLL, nLH, nHL, nHH, out, ws);

    // pass 3: recompute + normalize + store
    fsc_cipher<true><<<dim3(192, 16), 256, 0, stream>>>(
        img, nLL, nLH, nHL, nHH, out, ws);
}